// AttnDecoderRNN_57501022159605
// MI455X (gfx1250) — compile-verified
//
#include <hip/hip_runtime.h>
#include <math.h>

#define VOCAB 50257
#define H 1024
#define SEQ 512

typedef __attribute__((ext_vector_type(2))) float v2f;
typedef __attribute__((ext_vector_type(8))) float v8f;

#define GEMV_RELU  1
#define GEMV_ACCUM 2

// y[r] = dot(x, W[r,:]) + bias[r], 16 rows per wave via V_WMMA_F32_16X16X4_F32.
// A fragment = vector x replicated into all 16 M rows (all D rows identical).
// B fragment = 16 weight rows; fragment K-slots are remapped so each lane loads
// one contiguous float4 per 8-K step (half 0 -> mem K {k..k+3}, half 1 -> {k+4..k+7}).
// wmma#0 consumes {k,k+1,k+4,k+5}, wmma#1 consumes {k+2,k+3,k+6,k+7}; A uses the
// identical slot->K mapping, so the dot product is exact.
__global__ __launch_bounds__(256) void k_gemv_wmma(
    const float* __restrict__ W, const float* __restrict__ x,
    const float* __restrict__ bias, float* __restrict__ y,
    int rows, int K, int flags)
{
    const int lane = threadIdx.x & 31;
    const int wv   = threadIdx.x >> 5;                 // wave in block: 0..7
    const int v0   = (blockIdx.x * 8 + wv) << 4;       // first output row of this wave
    if (v0 >= rows) return;                            // wave-uniform: EXEC stays all-ones

    const int half = lane >> 4;                        // K-half select
    const int l16  = lane & 15;                        // N / output row within tile
    int row = v0 + l16;
    if (row >= rows) row = rows - 1;                   // clamp tail (result discarded)

    // float4 element index: 2*step + half  -> byte offset 32*step + 16*half
    const float4* wq = reinterpret_cast<const float4*>(W + (size_t)row * K) + half;
    const float4* xq = reinterpret_cast<const float4*>(x) + half;

    v8f acc0 = {0.f,0.f,0.f,0.f,0.f,0.f,0.f,0.f};
    v8f acc1 = {0.f,0.f,0.f,0.f,0.f,0.f,0.f,0.f};

    const int steps = K >> 3;                          // K is a multiple of 8
    for (int s = 0; s < steps; ++s) {
        float4 a = xq[2 * s];                          // broadcast within half (cache hit)
        float4 b = wq[2 * s];                          // one b128 per lane per 8 K
        v2f a0; a0.x = a.x; a0.y = a.y;
        v2f b0; b0.x = b.x; b0.y = b.y;
        v2f a1; a1.x = a.z; a1.y = a.w;
        v2f b1; b1.x = b.z; b1.y = b.w;
        acc0 = __builtin_amdgcn_wmma_f32_16x16x4_f32(false, a0, false, b0,
                                                     (short)0, acc0, false, false);
        acc1 = __builtin_amdgcn_wmma_f32_16x16x4_f32(false, a1, false, b1,
                                                     (short)0, acc1, false, false);
    }

    if (half == 0) {                                   // D row M=0 lives in lanes 0-15, acc[0]
        int orow = v0 + l16;
        if (orow < rows) {
            float val = acc0[0] + acc1[0] + (bias ? bias[orow] : 0.f);
            if (flags & GEMV_RELU)  val = fmaxf(val, 0.f);
            if (flags & GEMV_ACCUM) val += y[orow];
            y[orow] = val;
        }
    }
}

// embedding lookup + attn_in = concat(embedded, h0)
__global__ __launch_bounds__(256) void k_prepare(const int* __restrict__ idx,
    const float* __restrict__ emb, const float* __restrict__ h0,
    float* __restrict__ embedded, float* __restrict__ attn_in)
{
    int j = blockIdx.x * blockDim.x + threadIdx.x;
    if (j < H) {
        float e = emb[(size_t)idx[0] * H + j];
        embedded[j] = e;
        attn_in[j]  = e;
    } else if (j < 2 * H) {
        attn_in[j] = h0[j - H];
    }
}

__global__ __launch_bounds__(512) void k_softmax512(const float* __restrict__ logits,
    float* __restrict__ w_ws, float* __restrict__ w_out)
{
    __shared__ float red[512];
    int t = threadIdx.x;
    float v = logits[t];
    red[t] = v; __syncthreads();
    for (int s = 256; s > 0; s >>= 1) {
        if (t < s) red[t] = fmaxf(red[t], red[t + s]);
        __syncthreads();
    }
    float m = red[0]; __syncthreads();
    float e = expf(v - m);
    red[t] = e; __syncthreads();
    for (int s = 256; s > 0; s >>= 1) {
        if (t < s) red[t] += red[t + s];
        __syncthreads();
    }
    float wt = e / red[0];
    w_ws[t]  = wt;
    w_out[t] = wt;
}

// attn_applied = attn_weights @ enc ; comb_in = concat(embedded, attn_applied)
__global__ __launch_bounds__(256) void k_attn_apply(const float* __restrict__ w,
    const float* __restrict__ enc, const float* __restrict__ embedded,
    float* __restrict__ attn_applied, float* __restrict__ comb_in)
{
    int hh = blockIdx.x * blockDim.x + threadIdx.x;
    if (hh >= H) return;
    float s = 0.f;
    for (int i = 0; i < SEQ; ++i) s = fmaf(w[i], enc[(size_t)i * H + hh], s);
    attn_applied[hh] = s;
    comb_in[hh]      = embedded[hh];
    comb_in[H + hh]  = s;
}

__global__ __launch_bounds__(256) void k_lstm(const float* __restrict__ gates,
    const float* __restrict__ c0, float* __restrict__ h_ws,
    float* __restrict__ h_out, float* __restrict__ c_out)
{
    int j = blockIdx.x * blockDim.x + threadIdx.x;
    if (j >= H) return;
    float ig = 1.f / (1.f + expf(-gates[j]));
    float fg = 1.f / (1.f + expf(-gates[H + j]));
    float gg = tanhf(gates[2 * H + j]);
    float og = 1.f / (1.f + expf(-gates[3 * H + j]));
    float c  = fg * c0[j] + ig * gg;
    float h  = og * tanhf(c);
    h_ws[j] = h; h_out[j] = h; c_out[j] = c;
}

__global__ __launch_bounds__(1024) void k_logsoftmax(const float* __restrict__ logits,
    float* __restrict__ out)
{
    __shared__ float red[1024];
    int t = threadIdx.x;
    float m = -3.402823466e+38f;
    for (int v = t; v < VOCAB; v += 1024) m = fmaxf(m, logits[v]);
    red[t] = m; __syncthreads();
    for (int s = 512; s > 0; s >>= 1) {
        if (t < s) red[t] = fmaxf(red[t], red[t + s]);
        __syncthreads();
    }
    float M = red[0]; __syncthreads();
    float sum = 0.f;
    for (int v = t; v < VOCAB; v += 1024) sum += expf(logits[v] - M);
    red[t] = sum; __syncthreads();
    for (int s = 512; s > 0; s >>= 1) {
        if (t < s) red[t] += red[t + s];
        __syncthreads();
    }
    float lse = M + logf(red[0]);
    for (int v = t; v < VOCAB; v += 1024) out[v] = logits[v] - lse;
}

extern "C" void kernel_launch(void* const* d_in, const int* in_sizes, int n_in,
                              void* d_out, int out_size, void* d_ws, size_t ws_size,
                              hipStream_t stream)
{
    (void)in_sizes; (void)n_in; (void)out_size; (void)ws_size;

    const int*   idx    = (const int*)  d_in[0];
    const float* h0     = (const float*)d_in[1];
    const float* c0     = (const float*)d_in[2];
    const float* enc    = (const float*)d_in[3];
    const float* emb    = (const float*)d_in[4];
    const float* attn_W = (const float*)d_in[5];
    const float* attn_b = (const float*)d_in[6];
    const float* comb_W = (const float*)d_in[7];
    const float* comb_b = (const float*)d_in[8];
    const float* w_ih   = (const float*)d_in[9];
    const float* w_hh   = (const float*)d_in[10];
    const float* b_ih   = (const float*)d_in[11];
    const float* b_hh   = (const float*)d_in[12];
    const float* out_W  = (const float*)d_in[13];
    const float* out_b  = (const float*)d_in[14];

    float* out      = (float*)d_out;
    float* o_logits = out;                    // [VOCAB]
    float* o_h      = out + VOCAB;            // [H]
    float* o_c      = out + VOCAB + H;        // [H]
    float* o_attnw  = out + VOCAB + 2 * H;    // [SEQ]

    float* ws       = (float*)d_ws;
    float* embedded = ws;                     // H
    float* attn_in  = ws + 1024;              // 2H
    float* attn_log = ws + 3072;              // SEQ
    float* attn_wt  = ws + 3584;              // SEQ
    float* attn_app = ws + 4096;              // H
    float* comb_in  = ws + 5120;              // 2H
    float* xbuf     = ws + 7168;              // H
    float* gates    = ws + 8192;              // 4H
    float* hbuf     = ws + 12288;             // H
    float* logits   = ws + 13312;             // VOCAB

    k_prepare<<<8, 256, 0, stream>>>(idx, emb, h0, embedded, attn_in);
    // attn logits: 512 rows, K=2048  (32 waves -> 4 blocks)
    k_gemv_wmma<<<4, 256, 0, stream>>>(attn_W, attn_in, attn_b, attn_log, SEQ, 2 * H, 0);
    k_softmax512<<<1, 512, 0, stream>>>(attn_log, attn_wt, o_attnw);
    k_attn_apply<<<4, 256, 0, stream>>>(attn_wt, enc, embedded, attn_app, comb_in);
    // combine + relu: 1024 rows, K=2048
    k_gemv_wmma<<<8, 256, 0, stream>>>(comb_W, comb_in, comb_b, xbuf, H, 2 * H, GEMV_RELU);
    // LSTM gates: two GEMVs, second accumulates
    k_gemv_wmma<<<32, 256, 0, stream>>>(w_ih, xbuf, b_ih, gates, 4 * H, H, 0);
    k_gemv_wmma<<<32, 256, 0, stream>>>(w_hh, h0, b_hh, gates, 4 * H, H, GEMV_ACCUM);
    k_lstm<<<4, 256, 0, stream>>>(gates, c0, hbuf, o_h, o_c);
    // vocab projection: 50257 rows, K=1024 (dominant HBM stream, ~206 MB)
    int vblocks = (VOCAB + 127) / 128;
    k_gemv_wmma<<<vblocks, 256, 0, stream>>>(out_W, hbuf, out_b, logits, VOCAB, H, 0);
    k_logsoftmax<<<1, 1024, 0, stream>>>(logits, o_logits);
}